// FAE_GraphConv_77653008712167
// MI455X (gfx1250) — compile-verified
//
#include <hip/hip_runtime.h>

typedef __attribute__((ext_vector_type(2))) float v2f;
typedef __attribute__((ext_vector_type(8))) float v8f;

#define N_NODES 50000
#define N_EDGES 600000
#define IN_CH   128
#define H1_CH   64
#define H2_CH   32
#define N_TILES (N_NODES / 16)   // 3125, exact

// Guaranteed hardware f32 atomic add (non-returning, STOREcnt-tracked).
// GV addressing mode: 64-bit address in a VGPR pair, no SADDR.
static __device__ __forceinline__ void atomic_add_f32(float* p, float v) {
    asm volatile("global_atomic_add_f32 %0, %1, off"
                 :
                 : "v"(p), "v"(v)
                 : "memory");
}

// ---------------- zero workspace ----------------
__global__ void k_zero(float* __restrict__ p, size_t n) {
    size_t i = (size_t)blockIdx.x * blockDim.x + threadIdx.x;
    if (i < n) p[i] = 0.0f;
}

// ---------------- degree ----------------
__global__ void k_degree(const long long* __restrict__ ei, float* __restrict__ deg) {
    int e = blockIdx.x * blockDim.x + threadIdx.x;
    if (e >= N_EDGES) return;
    int d = (int)ei[N_EDGES + e];   // dst row
    atomic_add_f32(&deg[d], 1.0f);
}

// ---------------- scatter-add feat[src] into agg[dst] ----------------
// lc = log2(channels/4); 1<<lc threads per edge, float4 per thread.
__global__ void k_scatter(const float* __restrict__ feat, const long long* __restrict__ ei,
                          float* __restrict__ agg, int ch, int lc) {
    long long gid = (long long)blockIdx.x * blockDim.x + threadIdx.x;
    long long total = (long long)N_EDGES << lc;
    if (gid >= total) return;
    int e  = (int)(gid >> lc);
    int c4 = (int)(gid & ((1 << lc) - 1)) * 4;
    int s = (int)ei[e];
    int d = (int)ei[N_EDGES + e];
    const float4 v = *(const float4*)(feat + (size_t)s * ch + c4);
    float* o = agg + (size_t)d * ch + c4;
    atomic_add_f32(o + 0, v.x);
    atomic_add_f32(o + 1, v.y);
    atomic_add_f32(o + 2, v.z);
    atomic_add_f32(o + 3, v.w);
}

// ---------------- GraphConv GEMM via V_WMMA_F32_16X16X4_F32 ----------------
// out[16 nodes x OUT] = relu( (agg*inv_deg) @ W_rel + b + xin @ W_root )
// One wave per 16-out-channel tile. K stepped by 4.
// f32 A-frag (16x4): lanes 0-15 hold {K=k,k+1}, lanes 16-31 hold {K=k+2,k+3}, M=lane%16.
// f32 B-frag (4x16): lanes 0-15 hold {K=k,k+1}, lanes 16-31 hold {K=k+2,k+3}, N=lane%16.
// C/D (16x16): VGPR j -> M = j + (lane<16 ? 0 : 8), N = lane%16.
template <int KDIM, int OUT>
__global__ void k_graphconv(const float* __restrict__ xin, const float* __restrict__ agg,
                            const float* __restrict__ deg,
                            const float* __restrict__ W_rel, const float* __restrict__ b_rel,
                            const float* __restrict__ W_root, float* __restrict__ out) {
    const int tile  = blockIdx.x;
    const int wave  = threadIdx.x >> 5;   // out-channel tile
    const int lane  = threadIdx.x & 31;
    const int m     = lane & 15;
    const int khalf = lane >> 4;          // 0 or 1
    const int row   = tile * 16 + m;
    const int n0    = wave * 16;
    const float scale = 1.0f / fmaxf(deg[row], 1.0f);

    const float* aRow = agg + (size_t)row * KDIM;
    const float* xRow = xin + (size_t)row * KDIM;

    v8f acc = {};
    #pragma unroll
    for (int k = 0; k < KDIM; k += 4) {
        const int ka  = k + 2 * khalf;
        const int col = n0 + m;
        v2f a_rel, a_root, b_relf, b_rootf;
        a_rel.x  = aRow[ka]     * scale;
        a_rel.y  = aRow[ka + 1] * scale;
        a_root.x = xRow[ka];
        a_root.y = xRow[ka + 1];
        b_relf.x  = W_rel [(size_t)ka       * OUT + col];
        b_relf.y  = W_rel [(size_t)(ka + 1) * OUT + col];
        b_rootf.x = W_root[(size_t)ka       * OUT + col];
        b_rootf.y = W_root[(size_t)(ka + 1) * OUT + col];
        acc = __builtin_amdgcn_wmma_f32_16x16x4_f32(false, a_rel,  false, b_relf,
                                                    (short)0, acc, false, false);
        acc = __builtin_amdgcn_wmma_f32_16x16x4_f32(false, a_root, false, b_rootf,
                                                    (short)0, acc, false, false);
    }

    const int ncol  = n0 + (lane & 15);
    const int mbase = (lane < 16) ? 0 : 8;
    const float bias = b_rel[ncol];
    #pragma unroll
    for (int j = 0; j < 8; ++j) {
        const int node = tile * 16 + mbase + j;
        out[(size_t)node * OUT + ncol] = fmaxf(acc[j] + bias, 0.0f);
    }
}

// ---------------- final linear: [N,32] @ [32,1] + b ----------------
__global__ void k_final(const float* __restrict__ h2, const float* __restrict__ W_lin,
                        const float* __restrict__ b_lin, float* __restrict__ out) {
    int i = blockIdx.x * blockDim.x + threadIdx.x;
    if (i >= N_NODES) return;
    float s = b_lin[0];
    const float* r = h2 + (size_t)i * H2_CH;
    #pragma unroll
    for (int j = 0; j < H2_CH; ++j) s += r[j] * W_lin[j];
    out[i] = s;
}

extern "C" void kernel_launch(void* const* d_in, const int* in_sizes, int n_in,
                              void* d_out, int out_size, void* d_ws, size_t ws_size,
                              hipStream_t stream) {
    const float*     x       = (const float*)d_in[0];
    const long long* ei      = (const long long*)d_in[1];
    const float*     W1_rel  = (const float*)d_in[2];
    const float*     b1_rel  = (const float*)d_in[3];
    const float*     W1_root = (const float*)d_in[4];
    const float*     W2_rel  = (const float*)d_in[5];
    const float*     b2_rel  = (const float*)d_in[6];
    const float*     W2_root = (const float*)d_in[7];
    const float*     W_lin   = (const float*)d_in[8];
    const float*     b_lin   = (const float*)d_in[9];
    float* out = (float*)d_out;

    // workspace layout (floats): deg[N] | agg1[N*128] | agg2[N*64] | h1[N*64] | h2[N*32]
    float* ws   = (float*)d_ws;
    float* deg  = ws;
    float* agg1 = deg  + (size_t)N_NODES;
    float* agg2 = agg1 + (size_t)N_NODES * IN_CH;
    float* h1   = agg2 + (size_t)N_NODES * H1_CH;
    float* h2   = h1   + (size_t)N_NODES * H1_CH;

    // 1) zero deg + agg1 + agg2 (h1/h2 are fully overwritten)
    const size_t nz = (size_t)N_NODES * (1 + IN_CH + H1_CH);
    k_zero<<<(unsigned)((nz + 511) / 512), 512, 0, stream>>>(ws, nz);

    // 2) degrees
    k_degree<<<(N_EDGES + 255) / 256, 256, 0, stream>>>(ei, deg);

    // 3) scatter x[src] -> agg1[dst]  (32 threads/edge, float4 each)
    {
        long long t = (long long)N_EDGES * 32;
        k_scatter<<<(unsigned)((t + 255) / 256), 256, 0, stream>>>(x, ei, agg1, IN_CH, 5);
    }

    // 4) layer-1 GraphConv GEMM (K=128, OUT=64): 4 waves per 16-node tile
    k_graphconv<IN_CH, H1_CH><<<N_TILES, 128, 0, stream>>>(x, agg1, deg, W1_rel, b1_rel,
                                                           W1_root, h1);

    // 5) scatter h1[src] -> agg2[dst] (16 threads/edge)
    {
        long long t = (long long)N_EDGES * 16;
        k_scatter<<<(unsigned)((t + 255) / 256), 256, 0, stream>>>(h1, ei, agg2, H1_CH, 4);
    }

    // 6) layer-2 GraphConv GEMM (K=64, OUT=32): 2 waves per 16-node tile
    k_graphconv<H1_CH, H2_CH><<<N_TILES, 64, 0, stream>>>(h1, agg2, deg, W2_rel, b2_rel,
                                                          W2_root, h2);

    // 7) final linear head
    k_final<<<(N_NODES + 255) / 256, 256, 0, stream>>>(h2, W_lin, b_lin, out);
}